// MemoryDecoder_71150428225786
// MI455X (gfx1250) — compile-verified
//
#include <hip/hip_runtime.h>
#include <hip/hip_bf16.h>
#include <stdint.h>

// Problem constants (match the reference)
#define B_   2
#define H1_  54
#define W1_  120
#define H2_  54
#define W2_  120
#define ND_  81     // (2R+1)^2
#define UP_  8
#define HW1  (H1_*W1_)        // 6480
#define MAP  (H2_*W2_)        // 6480
#define QTOT (B_*HW1)         // 12960

// ---------------------------------------------------------------------------
// Kernel 1: correlation gather. One wave32 per query point (8 points/block,
// rem-consecutive within one batch image since 6480 % 8 == 0).
//  Phase 1: CDNA5 async global->LDS staging of the <=10x10 bilinear window
//           (ASYNCcnt, no VGPR round-trip), s_wait_asynccnt.
//  Phase 2: lanes split the 81 deltas, serve all 324 taps from LDS, deposit
//           results into a padded LDS transpose buffer (stride 9 -> conflict
//           free for stride-8-readers and stride-9-writers, gcd(9,64)=1).
//  Phase 3: block-wide coalesced store: 8 consecutive threads write 8
//           consecutive floats (full 32B sectors, no store amplification).
// ---------------------------------------------------------------------------
__global__ __launch_bounds__(256) void corr_kernel(const float* __restrict__ cost,
                                                   const float* __restrict__ coords,
                                                   float* __restrict__ out)
{
    __shared__ float tile[8][120];        // 8 waves x (10 rows * stride 12)
    __shared__ float lout[ND_ * 9];       // [t*9 + p], p = point-in-block
    const int wave = threadIdx.x >> 5;
    const int lane = threadIdx.x & 31;
    const int q    = blockIdx.x * 8 + wave;   // grid = QTOT/8 exactly

    const int b    = q / HW1;
    const int rem  = q - b * HW1;
    const int rem0 = rem & ~7;                // block-uniform

    const float cx = coords[(b * 2 + 0) * HW1 + rem];
    const float cy = coords[(b * 2 + 1) * HW1 + rem];
    const float fcx = floorf(cx);
    const float fcy = floorf(cy);
    const int   Xi  = (int)fcx;
    const int   Yi  = (int)fcy;
    const float fx  = cx - fcx;   // x = cx + integer delta -> same fraction
    const float fy  = cy - fcy;

    const float* qbase  = cost + (size_t)q * MAP;
    const unsigned ldsb = (unsigned)(uintptr_t)&tile[wave][0];

    // ---- Phase 1: async stage 10x10 window (clamped addrs; OOB taps get
    //      zero weight later, so clamped duplicates are harmless).
    #pragma unroll
    for (int r = 0; r < 4; ++r) {
        int e = lane + r * 32;
        e = (e < 100) ? e : 99;               // padding keeps EXEC all-ones
        const int ty = e / 10, tx = e - ty * 10;
        int gy = Yi - 4 + ty; gy = gy < 0 ? 0 : (gy > H2_ - 1 ? H2_ - 1 : gy);
        int gx = Xi - 4 + tx; gx = gx < 0 ? 0 : (gx > W2_ - 1 ? W2_ - 1 : gx);
        const unsigned long long ga =
            (unsigned long long)(uintptr_t)(qbase + gy * W2_ + gx);
        const unsigned la = ldsb + (unsigned)(ty * 12 + tx) * 4u;
        asm volatile("global_load_async_to_lds_b32 %0, %1, off"
                     :: "v"(la), "v"(ga) : "memory");
    }
    asm volatile("s_wait_asynccnt 0" ::: "memory");   // own-wave consumption

    // ---- Phase 2: 81 bilinear taps from LDS.
    const float w00 = (1.f - fx) * (1.f - fy);
    const float w10 = fx         * (1.f - fy);
    const float w01 = (1.f - fx) * fy;
    const float w11 = fx         * fy;

    for (int t = lane; t < ND_; t += 32) {
        const int ddy = t / 9 - 4;               // added to x (reference quirk)
        const int ddx = (t - (t / 9) * 9) - 4;   // added to y
        const int x0 = Xi + ddy;
        const int y0 = Yi + ddx;
        const int lx = ddy + 4;
        const int ly = ddx + 4;

        const bool vx0 = (x0     >= 0) & (x0     < W2_);
        const bool vx1 = (x0 + 1 >= 0) & (x0 + 1 < W2_);
        const bool vy0 = (y0     >= 0) & (y0     < H2_);
        const bool vy1 = (y0 + 1 >= 0) & (y0 + 1 < H2_);

        float s = 0.f;
        s += (vx0 & vy0) ? w00 * tile[wave][ ly      * 12 + lx    ] : 0.f;
        s += (vx1 & vy0) ? w10 * tile[wave][ ly      * 12 + lx + 1] : 0.f;
        s += (vx0 & vy1) ? w01 * tile[wave][(ly + 1) * 12 + lx    ] : 0.f;
        s += (vx1 & vy1) ? w11 * tile[wave][(ly + 1) * 12 + lx + 1] : 0.f;

        lout[t * 9 + wave] = s;
    }
    __syncthreads();

    // ---- Phase 3: coalesced block-wide store.
    float* obase = out + (size_t)b * ND_ * HW1 + rem0;
    for (int i = threadIdx.x; i < ND_ * 8; i += 256) {
        const int t = i >> 3;
        const int p = i & 7;
        obase[(size_t)t * HW1 + p] = lout[t * 9 + p];
    }
}

// ---------------------------------------------------------------------------
// Kernel 2: convex upsample. Thread per (n,h,w); fused 9-way softmax +
// weighted sum. Mask loads are lane-contiguous; outputs buffered per 8-j row
// and stored as two float4 (32B contiguous per thread -> global_store_b128).
// ---------------------------------------------------------------------------
__global__ __launch_bounds__(256) void upsample_kernel(const float* __restrict__ flow,
                                                       const float* __restrict__ mask,
                                                       float* __restrict__ out)
{
    const int idx = blockIdx.x * 256 + threadIdx.x;
    if (idx >= QTOT) return;
    const int n   = idx / HW1;
    const int rem = idx - n * HW1;
    const int h   = rem / W1_;
    const int w   = rem - h * W1_;

    float nbx[9], nby[9];
    #pragma unroll
    for (int k = 0; k < 9; ++k) {
        const int di = k / 3, dj = k - di * 3;
        const int hh = h + di - 1, ww = w + dj - 1;
        const bool v = (hh >= 0) & (hh < H1_) & (ww >= 0) & (ww < W1_);
        const int o  = hh * W1_ + ww;
        nbx[k] = v ? 8.f * flow[(size_t)(n * 2 + 0) * HW1 + o] : 0.f;
        nby[k] = v ? 8.f * flow[(size_t)(n * 2 + 1) * HW1 + o] : 0.f;
    }

    const float* mb = mask + (size_t)n * 576 * HW1 + rem;
    float* outx = out + ((size_t)(n * 2 + 0) * (H1_ * UP_) + (size_t)h * UP_) * (W1_ * UP_)
                      + (size_t)w * UP_;
    float* outy = out + ((size_t)(n * 2 + 1) * (H1_ * UP_) + (size_t)h * UP_) * (W1_ * UP_)
                      + (size_t)w * UP_;

    #pragma unroll 1
    for (int i = 0; i < 8; ++i) {
        float bufx[8], bufy[8];
        #pragma unroll
        for (int j = 0; j < 8; ++j) {
            const int ij = i * 8 + j;
            float m[9];
            float mx = -3.402823466e38f;
            #pragma unroll
            for (int k = 0; k < 9; ++k) {
                m[k] = mb[(size_t)(k * 64 + ij) * HW1];
                mx = fmaxf(mx, m[k]);
            }
            float sum = 0.f, ax = 0.f, ay = 0.f;
            #pragma unroll
            for (int k = 0; k < 9; ++k) {
                const float p = __expf(m[k] - mx);
                sum += p;
                ax  += p * nbx[k];
                ay  += p * nby[k];
            }
            const float inv = 1.f / sum;
            bufx[j] = ax * inv;
            bufy[j] = ay * inv;
        }
        // Two 16B stores per channel: contiguous, 32B-aligned per thread.
        float* rx = outx + (size_t)i * (W1_ * UP_);
        float* ry = outy + (size_t)i * (W1_ * UP_);
        *(float4*)(rx)     = *(const float4*)&bufx[0];
        *(float4*)(rx + 4) = *(const float4*)&bufx[4];
        *(float4*)(ry)     = *(const float4*)&bufy[0];
        *(float4*)(ry + 4) = *(const float4*)&bufy[4];
    }
}

// ---------------------------------------------------------------------------
extern "C" void kernel_launch(void* const* d_in, const int* in_sizes, int n_in,
                              void* d_out, int out_size, void* d_ws, size_t ws_size,
                              hipStream_t stream) {
    (void)in_sizes; (void)n_in; (void)out_size; (void)d_ws; (void)ws_size;
    const float* cost   = (const float*)d_in[0];   // (B*H1*W1, 1, 54, 120)
    const float* coords = (const float*)d_in[1];   // (B, 2, 54, 120)  [cx, cy]
    const float* flow   = (const float*)d_in[2];   // (B, 2, 54, 120)
    const float* mask   = (const float*)d_in[3];   // (B, 576, 54, 120)

    float* out_corr = (float*)d_out;                               // (2,81,54,120)
    float* out_up   = out_corr + (size_t)B_ * ND_ * H1_ * W1_;     // (2,2,432,960)

    corr_kernel<<<QTOT / 8, 256, 0, stream>>>(cost, coords, out_corr);
    upsample_kernel<<<(QTOT + 255) / 256, 256, 0, stream>>>(flow, mask, out_up);
}